// TextureConsistencyLoss_6468220748288
// MI455X (gfx1250) — compile-verified
//
#include <hip/hip_runtime.h>
#include <hip/hip_bf16.h>

typedef float v2f __attribute__((ext_vector_type(2)));
typedef float v8f __attribute__((ext_vector_type(8)));

#define HH 384
#define WW 384
#define HW (HH * WW)
#define OH 768
#define OW 768

// ---------------------------------------------------------------------------
// General 4x4 inverse (cofactor expansion), row-major.
// ---------------------------------------------------------------------------
__device__ inline void invert4x4(const float m[16], float out[16]) {
    float inv[16];
    inv[0]  =  m[5]*m[10]*m[15] - m[5]*m[11]*m[14] - m[9]*m[6]*m[15] + m[9]*m[7]*m[14] + m[13]*m[6]*m[11] - m[13]*m[7]*m[10];
    inv[4]  = -m[4]*m[10]*m[15] + m[4]*m[11]*m[14] + m[8]*m[6]*m[15] - m[8]*m[7]*m[14] - m[12]*m[6]*m[11] + m[12]*m[7]*m[10];
    inv[8]  =  m[4]*m[9]*m[15]  - m[4]*m[11]*m[13] - m[8]*m[5]*m[15] + m[8]*m[7]*m[13] + m[12]*m[5]*m[11] - m[12]*m[7]*m[9];
    inv[12] = -m[4]*m[9]*m[14]  + m[4]*m[10]*m[13] + m[8]*m[5]*m[14] - m[8]*m[6]*m[13] - m[12]*m[5]*m[10] + m[12]*m[6]*m[9];
    inv[1]  = -m[1]*m[10]*m[15] + m[1]*m[11]*m[14] + m[9]*m[2]*m[15] - m[9]*m[3]*m[14] - m[13]*m[2]*m[11] + m[13]*m[3]*m[10];
    inv[5]  =  m[0]*m[10]*m[15] - m[0]*m[11]*m[14] - m[8]*m[2]*m[15] + m[8]*m[3]*m[14] + m[12]*m[2]*m[11] - m[12]*m[3]*m[10];
    inv[9]  = -m[0]*m[9]*m[15]  + m[0]*m[11]*m[13] + m[8]*m[1]*m[15] - m[8]*m[3]*m[13] - m[12]*m[1]*m[11] + m[12]*m[3]*m[9];
    inv[13] =  m[0]*m[9]*m[14]  - m[0]*m[10]*m[13] - m[8]*m[1]*m[14] + m[8]*m[2]*m[13] + m[12]*m[1]*m[10] - m[12]*m[2]*m[9];
    inv[2]  =  m[1]*m[6]*m[15]  - m[1]*m[7]*m[14]  - m[5]*m[2]*m[15] + m[5]*m[3]*m[14] + m[13]*m[2]*m[7]  - m[13]*m[3]*m[6];
    inv[6]  = -m[0]*m[6]*m[15]  + m[0]*m[7]*m[14]  + m[4]*m[2]*m[15] - m[4]*m[3]*m[14] - m[12]*m[2]*m[7]  + m[12]*m[3]*m[6];
    inv[10] =  m[0]*m[5]*m[15]  - m[0]*m[7]*m[13]  - m[4]*m[1]*m[15] + m[4]*m[3]*m[13] + m[12]*m[1]*m[7]  - m[12]*m[3]*m[5];
    inv[14] = -m[0]*m[5]*m[14]  + m[0]*m[6]*m[13]  + m[4]*m[1]*m[14] - m[4]*m[2]*m[13] - m[12]*m[1]*m[6]  + m[12]*m[2]*m[5];
    inv[3]  = -m[1]*m[6]*m[11]  + m[1]*m[7]*m[10]  + m[5]*m[2]*m[11] - m[5]*m[3]*m[10] - m[9]*m[2]*m[7]   + m[9]*m[3]*m[6];
    inv[7]  =  m[0]*m[6]*m[11]  - m[0]*m[7]*m[10]  - m[4]*m[2]*m[11] + m[4]*m[3]*m[10] + m[8]*m[2]*m[7]   - m[8]*m[3]*m[6];
    inv[11] = -m[0]*m[5]*m[11]  + m[0]*m[7]*m[9]   + m[4]*m[1]*m[11] - m[4]*m[3]*m[9]  - m[8]*m[1]*m[7]   + m[8]*m[3]*m[5];
    inv[15] =  m[0]*m[5]*m[10]  - m[0]*m[6]*m[9]   - m[4]*m[1]*m[10] + m[4]*m[2]*m[9]  + m[8]*m[1]*m[6]   - m[8]*m[2]*m[5];
    float det = m[0]*inv[0] + m[1]*inv[4] + m[2]*inv[8] + m[3]*inv[12];
    det = 1.0f / det;
    for (int i = 0; i < 16; ++i) out[i] = inv[i] * det;
}

// ---------------------------------------------------------------------------
// Kernel 1: per-batch M_b = (K * dst_RT) * inv(K * src_RT)
// One 32-lane wave per batch. The final 4x4 x 4x4 product is done with
// V_WMMA_F32_16X16X4_F32 (A = P2 padded to 16x4, B = P1inv padded to 4x16).
// ---------------------------------------------------------------------------
__global__ void k_compose(const float* __restrict__ K,
                          const float* __restrict__ srcRT,
                          const float* __restrict__ dstRT,
                          float* __restrict__ Mout, int nv) {
    int b = blockIdx.x;     // 0..B-1
    int s = b / nv;
    int lane = threadIdx.x; // 0..31
    __shared__ float P2s[16];   // K * dst_RT
    __shared__ float P1i[16];   // inv(K * src_RT)

    if (lane == 0) {
        float Km[16], A1[16], A2[16], P1[16], P2[16], inv[16];
        for (int t = 0; t < 16; ++t) {
            Km[t] = K[s * 16 + t];
            A1[t] = srcRT[b * 16 + t];
            A2[t] = dstRT[s * 16 + t];
        }
        for (int r = 0; r < 4; ++r)
            for (int c = 0; c < 4; ++c) {
                float a1 = 0.f, a2 = 0.f;
                for (int k = 0; k < 4; ++k) {
                    a1 += Km[r * 4 + k] * A1[k * 4 + c];
                    a2 += Km[r * 4 + k] * A2[k * 4 + c];
                }
                P1[r * 4 + c] = a1;
                P2[r * 4 + c] = a2;
            }
        invert4x4(P1, inv);
        for (int t = 0; t < 16; ++t) { P2s[t] = P2[t]; P1i[t] = inv[t]; }
    }
    __syncthreads();

    int r  = lane & 15;
    int hi = lane >> 4;
    // A-matrix 16x4 f32 layout: VGPR0 = K{0|2}, VGPR1 = K{1|3} (lo|hi half-wave)
    v2f a, bm;
    a.x  = (r < 4) ? P2s[r * 4 + (hi ? 2 : 0)] : 0.f;
    a.y  = (r < 4) ? P2s[r * 4 + (hi ? 3 : 1)] : 0.f;
    // B-matrix 4x16 f32 layout: lo lanes carry K rows 0..1, hi lanes rows 2..3
    bm.x = (r < 4) ? P1i[(hi ? 2 : 0) * 4 + r] : 0.f;
    bm.y = (r < 4) ? P1i[(hi ? 3 : 1) * 4 + r] : 0.f;

    v8f c = {};
    c = __builtin_amdgcn_wmma_f32_16x16x4_f32(false, a, false, bm,
                                              (short)0, c, false, false);
    // D layout: lane n (0..15), VGPR v -> D[v][n]. M = D[0..3][0..3].
    if (!hi && r < 4) {
#pragma unroll
        for (int v = 0; v < 4; ++v) Mout[b * 16 + v * 4 + r] = c[v];
    }
}

// ---------------------------------------------------------------------------
// Kernel 2: bilinear 768->384 downscale == exact 2x2 box filter
// (half-pixel centers, scale 2: src coord = 2*o + 0.5).
// ---------------------------------------------------------------------------
__global__ void k_resize(const float* __restrict__ dst, float* __restrict__ out,
                         int total) {
    int o = blockIdx.x * blockDim.x + threadIdx.x;
    if (o >= total) return;
    int j  = o % WW;
    int i  = (o / WW) % HH;
    int sc = o / HW;                 // (s*3 + c)
    const float* ip = dst + (size_t)sc * OH * OW;
    int r0 = 2 * i, c0 = 2 * j;
    float v = ip[(size_t)r0 * OW + c0]     + ip[(size_t)r0 * OW + c0 + 1]
            + ip[(size_t)(r0+1) * OW + c0] + ip[(size_t)(r0+1) * OW + c0 + 1];
    out[o] = 0.25f * v;
}

// ---------------------------------------------------------------------------
// Kernel 3: fused project + grid-sample + masked L1, block-reduced, atomics.
// ---------------------------------------------------------------------------
__global__ void k_loss(const float* __restrict__ pred_pts,
                       const float* __restrict__ src_colors,
                       const float* __restrict__ Mbuf,
                       const float* __restrict__ resized,
                       float* __restrict__ acc, int nv) {
    const int b   = blockIdx.y;
    const int pix = blockIdx.x * blockDim.x + threadIdx.x;
    const int t   = threadIdx.x;

    float num = 0.f, cnt = 0.f;
    if (pix < HW) {
        const int i = pix / WW;
        const int j = pix % WW;
        const float d = pred_pts[(size_t)b * HW + pix];
        const float* M = Mbuf + b * 16;
        const float x = (float)j * d, y = (float)i * d, z = d;
        const float p0 = M[0] * x + M[1] * y + M[2]  * z + M[3];
        const float p1 = M[4] * x + M[5] * y + M[6]  * z + M[7];
        const float p2 = M[8] * x + M[9] * y + M[10] * z + M[11];
        const float iz = 1.0f / (p2 + 1e-7f);
        const float px = p0 * iz * (1.0f / ((WW - 1) * 0.5f)) - 1.0f;
        const float py = p1 * iz * (1.0f / ((HH - 1) * 0.5f)) - 1.0f;
        const bool valid = (px >= -1.f) && (px <= 1.f) && (py >= -1.f) && (py <= 1.f);

        if (valid) {
            const float ix = ((px + 1.0f) * WW - 1.0f) * 0.5f;
            const float iy = ((py + 1.0f) * HH - 1.0f) * 0.5f;
            const float x0 = floorf(ix), y0 = floorf(iy);
            const float x1 = x0 + 1.0f,  y1 = y0 + 1.0f;
            const float wx1 = ix - x0, wx0 = 1.0f - wx1;
            const float wy1 = iy - y0, wy0 = 1.0f - wy1;

            const int s = b / nv;
            const float* img = resized + (size_t)s * 3 * HW;

            float w0 = 0.f, w1 = 0.f, w2 = 0.f;
#pragma unroll
            for (int corner = 0; corner < 4; ++corner) {
                const float xf = (corner & 1) ? x1 : x0;
                const float yf = (corner & 2) ? y1 : y0;
                const float wx = (corner & 1) ? wx1 : wx0;
                const float wy = (corner & 2) ? wy1 : wy0;
                const float wgt = wx * wy;
                const float inb = (xf >= 0.f && xf <= (float)(WW - 1) &&
                                   yf >= 0.f && yf <= (float)(HH - 1)) ? 1.f : 0.f;
                const int xc = (int)fminf(fmaxf(xf, 0.f), (float)(WW - 1));
                const int yc = (int)fminf(fmaxf(yf, 0.f), (float)(HH - 1));
                const int idx = yc * WW + xc;
                const float ww = wgt * inb;
                w0 += ww * img[idx];
                w1 += ww * img[HW + idx];
                w2 += ww * img[2 * HW + idx];
            }
            const size_t sb = (size_t)b * 3 * HW + pix;
            const float s0 = src_colors[sb];
            const float s1 = src_colors[sb + HW];
            const float s2 = src_colors[sb + 2 * HW];
            num = fabsf(w0 - s0) + fabsf(w1 - s1) + fabsf(w2 - s2);
            cnt = 1.f;
        }
    }

    __shared__ float sn[256];
    __shared__ float sc2[256];
    sn[t] = num;
    sc2[t] = cnt;
    __syncthreads();
    for (int off = 128; off > 0; off >>= 1) {
        if (t < off) { sn[t] += sn[t + off]; sc2[t] += sc2[t + off]; }
        __syncthreads();
    }
    if (t == 0) {
        atomicAdd(&acc[0], sn[0]);
        atomicAdd(&acc[1], sc2[0]);
    }
}

__global__ void k_final(const float* __restrict__ acc, float* __restrict__ out) {
    out[0] = acc[0] / (3.0f * fmaxf(acc[1], 1.0f));
}

// ---------------------------------------------------------------------------
extern "C" void kernel_launch(void* const* d_in, const int* in_sizes, int n_in,
                              void* d_out, int out_size, void* d_ws, size_t ws_size,
                              hipStream_t stream) {
    (void)n_in; (void)out_size; (void)ws_size;
    const float* pred_pts   = (const float*)d_in[0];
    const float* src_colors = (const float*)d_in[1];
    const float* dst_colors = (const float*)d_in[2];
    const float* Kmat       = (const float*)d_in[3];
    const float* inRT       = (const float*)d_in[4];
    const float* outRT      = (const float*)d_in[5];
    float* out = (float*)d_out;

    const int ns   = in_sizes[3] / 16;   // K is (ns,4,4)
    const int Btot = in_sizes[4] / 16;   // input_RTs is (ns,nv,4,4)
    const int nv   = Btot / ns;

    char*  ws      = (char*)d_ws;
    float* acc     = (float*)ws;                       // 2 floats (+pad)
    float* Mbuf    = (float*)(ws + 256);               // Btot*16 floats
    float* resized = (float*)(ws + 256 + ((Btot * 16 * 4 + 255) / 256) * 256);

    hipMemsetAsync(acc, 0, 2 * sizeof(float), stream);

    k_compose<<<Btot, 32, 0, stream>>>(Kmat, inRT, outRT, Mbuf, nv);

    const int rTot = ns * 3 * HW;
    k_resize<<<(rTot + 255) / 256, 256, 0, stream>>>(dst_colors, resized, rTot);

    dim3 grid((HW + 255) / 256, Btot);
    k_loss<<<grid, 256, 0, stream>>>(pred_pts, src_colors, Mbuf, resized, acc, nv);

    k_final<<<1, 1, 0, stream>>>(acc, out);
}